// IpaBlock_56135222559300
// MI455X (gfx1250) — compile-verified
//
#include <hip/hip_runtime.h>
#include <cmath>

#define HID   128
#define HEADS 8
#define HD    16

typedef float v2f __attribute__((ext_vector_type(2)));
typedef float v8f __attribute__((ext_vector_type(8)));

// ---------------------------------------------------------------------
// zero fill
// ---------------------------------------------------------------------
__global__ void zero_kernel(unsigned* __restrict__ p, long n) {
    long i = (long)blockIdx.x * blockDim.x + threadIdx.x;
    if (i < n) p[i] = 0u;
}

// ---------------------------------------------------------------------
// 128x128 transpose: WT[n][k] = W[k][n]  (so GEMM B loads are contiguous)
// ---------------------------------------------------------------------
__global__ void transpose128_kernel(const float* __restrict__ W, float* __restrict__ WT) {
    int i = blockIdx.x * blockDim.x + threadIdx.x;
    if (i < HID * HID) {
        int r = i >> 7, c = i & 127;
        WT[(size_t)c * HID + r] = W[(size_t)r * HID + c];
    }
}

// ---------------------------------------------------------------------
// GEMM: out[nrows x 128] = A[nrows x 128] @ W[128 x 128] + bias
//       (+ residual) (+ SiLU).  W passed TRANSPOSED (WT[n][k]).
// One wave per 16x16 tile via V_WMMA_F32_16X16X4_F32.
// Block = 256 thr = 8 waves = the 8 column tiles of one 16-row band.
// A tile (16x128 = 8KB) staged in LDS once per block; B from WT as b64.
// ---------------------------------------------------------------------
__global__ void gemm_wmma_f32(const float* __restrict__ A,
                              const float* __restrict__ WT,
                              const float* __restrict__ bias,
                              const float* __restrict__ residual,
                              float* __restrict__ out,
                              int act /*0=none,1=silu*/) {
    __shared__ float As[16 * HID];                       // 8 KB
    const int rowTile = blockIdx.x;
    {   // cooperative b128 stage of the A tile
        const float4* s4 = (const float4*)(A + (size_t)rowTile * 16 * HID);
        float4* d4 = (float4*)As;
        for (int i = threadIdx.x; i < 16 * HID / 4; i += blockDim.x) d4[i] = s4[i];
    }
    __syncthreads();

    const int lane    = threadIdx.x & 31;
    const int colTile = threadIdx.x >> 5;                // 0..7
    const int m    = lane & 15;                          // row (A) / col (B) in tile
    const int khi  = lane >> 4;                          // 0 or 1
    const int koff = khi << 1;                           // 0 or 2
    const float* WTn = WT + (size_t)(colTile * 16 + m) * HID;

    v8f acc = {0.f, 0.f, 0.f, 0.f, 0.f, 0.f, 0.f, 0.f};
#pragma unroll
    for (int k = 0; k < HID; k += 4) {
        v2f a = *(const v2f*)&As[m * HID + k + koff];    // ds_load_b64
        v2f b = *(const v2f*)&WTn[k + koff];             // global_load_b64
        acc = __builtin_amdgcn_wmma_f32_16x16x4_f32(
            false, a, false, b, (short)0, acc, false, false);
    }

    const int   n  = colTile * 16 + m;
    const float bn = bias[n];
#pragma unroll
    for (int r = 0; r < 8; ++r) {
        int   row = rowTile * 16 + (khi << 3) + r;
        float val = acc[r] + bn;
        if (residual) val += residual[(size_t)row * HID + n];
        if (act) val = val / (1.0f + __expf(-val));      // SiLU
        out[(size_t)row * HID + n] = val;
    }
}

// ---------------------------------------------------------------------
// CSR build: degree histogram -> single-block scan -> bucket fill
// ---------------------------------------------------------------------
__global__ void degree_kernel(const int* __restrict__ src, unsigned* __restrict__ deg, int E_) {
    int e = blockIdx.x * blockDim.x + threadIdx.x;
    if (e < E_) atomicAdd(&deg[src[e]], 1u);
}

#define SCAN_T 1024
__global__ void scan_kernel(const unsigned* __restrict__ deg,
                            unsigned* __restrict__ rowptr,
                            unsigned* __restrict__ cursor, int n) {
    __shared__ unsigned sums[SCAN_T];
    const int t = threadIdx.x;
    const int chunk = (n + SCAN_T - 1) / SCAN_T;
    const int b0 = t * chunk;
    const int b1 = min(b0 + chunk, n);
    unsigned s = 0;
    for (int i = b0; i < b1; ++i) s += deg[i];
    sums[t] = s;
    __syncthreads();
    for (int off = 1; off < SCAN_T; off <<= 1) {         // inclusive scan
        unsigned v = (t >= off) ? sums[t - off] : 0u;
        __syncthreads();
        sums[t] += v;
        __syncthreads();
    }
    unsigned pre = (t == 0) ? 0u : sums[t - 1];
    for (int i = b0; i < b1; ++i) {
        rowptr[i] = pre;
        cursor[i] = pre;
        pre += deg[i];
    }
    if (b0 < n && b1 == n) rowptr[n] = pre;
}

__global__ void fill_kernel(const int* __restrict__ src, unsigned* __restrict__ cursor,
                            unsigned* __restrict__ eid, int E_) {
    int e = blockIdx.x * blockDim.x + threadIdx.x;
    if (e < E_) {
        unsigned pos = atomicAdd(&cursor[src[e]], 1u);
        eid[pos] = (unsigned)e;
    }
}

// ---------------------------------------------------------------------
// per (edge, head): raw logits + 6 edge biases (no atomics; CSR pass
// handles max/softmax per node)
// ---------------------------------------------------------------------
__global__ void edge_logits_kernel(const float* __restrict__ q,
                                   const float* __restrict__ k,
                                   const int* __restrict__ src,
                                   const int* __restrict__ dst,
                                   const float* __restrict__ dist,
                                   const float* __restrict__ bpp,
                                   const float* __restrict__ msa,
                                   const float* __restrict__ chem,
                                   const float* __restrict__ rel,
                                   const float* __restrict__ chain,
                                   const float* __restrict__ w_dist, const float* __restrict__ b_dist,
                                   const float* __restrict__ w_bpp,  const float* __restrict__ b_bpp,
                                   const float* __restrict__ w_msa,  const float* __restrict__ b_msa,
                                   const float* __restrict__ w_chem, const float* __restrict__ b_chem,
                                   const float* __restrict__ w_rel,  const float* __restrict__ b_rel,
                                   const float* __restrict__ w_chain,const float* __restrict__ b_chain,
                                   float* __restrict__ lg, int E_) {
    int i = blockIdx.x * blockDim.x + threadIdx.x;
    if (i >= E_ * HEADS) return;
    int e = i >> 3, hh = i & 7;
    int s = src[e], d = dst[e];

    const float4* qp = (const float4*)(q + (size_t)s * HID + hh * HD);
    const float4* kp = (const float4*)(k + (size_t)d * HID + hh * HD);
    float acc = 0.f;
#pragma unroll
    for (int j = 0; j < HD / 4; ++j) {
        float4 a = qp[j], b = kp[j];
        acc += a.x * b.x + a.y * b.y + a.z * b.z + a.w * b.w;
    }
    float logit = acc * 0.25f;                 // 1/sqrt(16)
    float dv = dist[e];
    logit -= dv * dv * w_dist[hh] + b_dist[hh];
    logit += bpp[e]  * w_bpp[hh]  + b_bpp[hh];
    logit += msa[e]  * w_msa[hh]  + b_msa[hh];
    logit += chem[e] * w_chem[hh] + b_chem[hh];
    logit += rel[e]  * w_rel[hh]  + b_rel[hh];
    logit += chain[e]* w_chain[hh]+ b_chain[hh];
    lg[i] = logit;
}

// ---------------------------------------------------------------------
// One wave per node, zero atomics:
//  phase 1: online softmax stats; lanes = 8 heads x 4 sub-lanes,
//           merged with shfl_xor(8), shfl_xor(16)
//  phase 2: one edge per iteration; lane j accumulates components
//           [4j..4j+3] (head j>>2); head-mean weight via shfl_xor 4/8/16
// Writes hup[node] (float4/lane) and disp[node] directly.
// ---------------------------------------------------------------------
__global__ void node_attn_kernel(const unsigned* __restrict__ rowptr,
                                 const unsigned* __restrict__ eid,
                                 const int* __restrict__ dst,
                                 const float* __restrict__ lg,
                                 const float* __restrict__ v,
                                 const float* __restrict__ x,
                                 float* __restrict__ hup,
                                 float* __restrict__ disp,
                                 int N_) {
    const int node = blockIdx.x * (blockDim.x >> 5) + (threadIdx.x >> 5);
    if (node >= N_) return;                       // wave-uniform
    const int lane = threadIdx.x & 31;
    const unsigned beg = rowptr[node], end = rowptr[node + 1];

    // ---- phase 1: per-head max & exp-sum (online) ----
    const int h = lane & 7, sub = lane >> 3;
    float mx = -3.0e38f, ssum = 0.f;
    for (unsigned i = beg + sub; i < end; i += 4) {
        unsigned e = eid[i];
        float l = lg[(size_t)e * HEADS + h];
        float mn = fmaxf(mx, l);
        ssum = ssum * __expf(mx - mn) + __expf(l - mn);
        mx = mn;
    }
    for (int msk = 8; msk <= 16; msk <<= 1) {     // merge the 4 sub-lanes
        float mo = __shfl_xor(mx, msk, 32);
        float so = __shfl_xor(ssum, msk, 32);
        float mn = fmaxf(mx, mo);
        ssum = ssum * __expf(mx - mn) + so * __expf(mo - mn);
        mx = mn;
    }
    // lanes 0..7 now hold final stats for heads 0..7; lane j wants head j>>2
    const int myhead = lane >> 2;
    const float m_my   = __shfl(mx, myhead, 32);
    const float inv_d  = 1.f / fmaxf(__shfl(ssum, myhead, 32), 1e-9f);

    // ---- phase 2: aggregate v and displacement ----
    float4 acc = {0.f, 0.f, 0.f, 0.f};
    float dacc = 0.f;
    const float xs = (lane < 3) ? x[(size_t)node * 3 + lane] : 0.f;
    for (unsigned i = beg; i < end; ++i) {
        unsigned e = eid[i];
        int d = dst[e];
        float w = __expf(lg[(size_t)e * HEADS + myhead] - m_my) * inv_d;
        float4 vv = *(const float4*)(v + (size_t)d * HID + lane * 4);
        acc.x += vv.x * w; acc.y += vv.y * w; acc.z += vv.z * w; acc.w += vv.w * w;
        // mean over heads: lanes {c, c+4, ..., c+28} cover each head once
        float wsum = w;
        wsum += __shfl_xor(wsum, 4, 32);
        wsum += __shfl_xor(wsum, 8, 32);
        wsum += __shfl_xor(wsum, 16, 32);
        if (lane < 3) dacc += (x[(size_t)d * 3 + lane] - xs) * (wsum * 0.125f);
    }
    *(float4*)(hup + (size_t)node * HID + lane * 4) = acc;
    if (lane < 3) disp[(size_t)node * 3 + lane] = dacc;
}

// ---------------------------------------------------------------------
// gate = tanh(t . wg2 + bg2);  x_out = x + gate * disp
// ---------------------------------------------------------------------
__global__ void gate_coords_kernel(const float* __restrict__ t,
                                   const float* __restrict__ wg2,
                                   const float* __restrict__ bg2,
                                   const float* __restrict__ x,
                                   const float* __restrict__ disp,
                                   float* __restrict__ xout,
                                   int n) {
    int i = blockIdx.x * blockDim.x + threadIdx.x;
    if (i >= n) return;
    const float4* tp = (const float4*)(t + (size_t)i * HID);
    const float4* wp = (const float4*)wg2;
    float acc = 0.f;
#pragma unroll
    for (int j = 0; j < HID / 4; ++j) {
        float4 a = tp[j], b = wp[j];
        acc += a.x * b.x + a.y * b.y + a.z * b.z + a.w * b.w;
    }
    float g = tanhf(acc + bg2[0]);
#pragma unroll
    for (int c = 0; c < 3; ++c)
        xout[(size_t)i * 3 + c] = x[(size_t)i * 3 + c] + g * disp[(size_t)i * 3 + c];
}

// ---------------------------------------------------------------------
extern "C" void kernel_launch(void* const* d_in, const int* in_sizes, int n_in,
                              void* d_out, int out_size, void* d_ws, size_t ws_size,
                              hipStream_t stream) {
    const int N = in_sizes[0] / HID;
    const int E = in_sizes[2];

    const float* h     = (const float*)d_in[0];
    const float* x     = (const float*)d_in[1];
    const int*   src   = (const int*)d_in[2];
    const int*   dst   = (const int*)d_in[3];
    const float* dist  = (const float*)d_in[4];
    const float* bpp   = (const float*)d_in[5];
    const float* msa   = (const float*)d_in[6];
    const float* chem  = (const float*)d_in[7];
    const float* rel   = (const float*)d_in[8];
    const float* chain = (const float*)d_in[9];
    const float* Wq = (const float*)d_in[10];
    const float* Wk = (const float*)d_in[11];
    const float* Wv = (const float*)d_in[12];
    const float* Wo = (const float*)d_in[13];
    const float* bq = (const float*)d_in[14];
    const float* bk = (const float*)d_in[15];
    const float* bv = (const float*)d_in[16];
    const float* bo = (const float*)d_in[17];
    const float* w_dist = (const float*)d_in[18]; const float* b_dist = (const float*)d_in[19];
    const float* w_bpp  = (const float*)d_in[20]; const float* b_bpp  = (const float*)d_in[21];
    const float* w_msa  = (const float*)d_in[22]; const float* b_msa  = (const float*)d_in[23];
    const float* w_chem = (const float*)d_in[24]; const float* b_chem = (const float*)d_in[25];
    const float* w_rel  = (const float*)d_in[26]; const float* b_rel  = (const float*)d_in[27];
    const float* w_chain= (const float*)d_in[28]; const float* b_chain= (const float*)d_in[29];
    const float* Wg1 = (const float*)d_in[30];
    const float* bg1 = (const float*)d_in[31];
    const float* wg2 = (const float*)d_in[32];
    const float* bg2 = (const float*)d_in[33];

    float* h_out = (float*)d_out;              // [N, 128]
    float* x_out = h_out + (size_t)N * HID;    // [N, 3]

    // ---- workspace layout ----
    float* ws = (float*)d_ws;
    float*    q     = ws;                                   // N*128 (reused as t)
    float*    kbuf  = q    + (size_t)N * HID;               // N*128 (reused as h_update)
    float*    vbuf  = kbuf + (size_t)N * HID;               // N*128
    float*    lg    = vbuf + (size_t)N * HID;               // E*8 logits
    float*    disp  = lg   + (size_t)E * HEADS;             // N*3
    unsigned* deg   = (unsigned*)(disp + (size_t)N * 3);    // N
    unsigned* rowp  = deg  + N;                             // N+1
    unsigned* curs  = rowp + N + 1;                         // N
    unsigned* eid   = curs + N;                             // E
    float*    WT    = (float*)(eid + E);                    // 5 * 128*128
    float* WqT  = WT;
    float* WkT  = WqT + HID * HID;
    float* WvT  = WkT + HID * HID;
    float* WoT  = WvT + HID * HID;
    float* Wg1T = WoT + HID * HID;
    float* hup = kbuf;                                      // alias (k dead after logits)
    float* t   = q;                                         // alias (q dead after logits)

    const int TB = 256;
    const int rowTiles = N / 16;                            // 20000 -> 1250
    const int wBlk = (HID * HID + TB - 1) / TB;             // 64

    // 0) transpose the five weight matrices (contiguous b64 B loads in GEMM)
    transpose128_kernel<<<wBlk, TB, 0, stream>>>(Wq,  WqT);
    transpose128_kernel<<<wBlk, TB, 0, stream>>>(Wk,  WkT);
    transpose128_kernel<<<wBlk, TB, 0, stream>>>(Wv,  WvT);
    transpose128_kernel<<<wBlk, TB, 0, stream>>>(Wo,  WoT);
    transpose128_kernel<<<wBlk, TB, 0, stream>>>(Wg1, Wg1T);

    // 1) CSR build: deg -> scan -> fill
    zero_kernel<<<(N + TB - 1) / TB, TB, 0, stream>>>(deg, N);
    degree_kernel<<<(E + TB - 1) / TB, TB, 0, stream>>>(src, deg, E);
    scan_kernel<<<1, SCAN_T, 0, stream>>>(deg, rowp, curs, N);
    fill_kernel<<<(E + TB - 1) / TB, TB, 0, stream>>>(src, curs, eid, E);

    // 2) Q/K/V projections (WMMA f32 16x16x4, LDS-staged A)
    gemm_wmma_f32<<<rowTiles, TB, 0, stream>>>(h, WqT, bq, nullptr, q,    0);
    gemm_wmma_f32<<<rowTiles, TB, 0, stream>>>(h, WkT, bk, nullptr, kbuf, 0);
    gemm_wmma_f32<<<rowTiles, TB, 0, stream>>>(h, WvT, bv, nullptr, vbuf, 0);

    // 3) raw edge logits
    {
        long nt = (long)E * HEADS;
        edge_logits_kernel<<<(nt + TB - 1) / TB, TB, 0, stream>>>(
            q, kbuf, src, dst, dist, bpp, msa, chem, rel, chain,
            w_dist, b_dist, w_bpp, b_bpp, w_msa, b_msa,
            w_chem, b_chem, w_rel, b_rel, w_chain, b_chain,
            lg, E);
    }

    // 4) per-node softmax + aggregation (wave per node, zero atomics)
    node_attn_kernel<<<(N + 7) / 8, TB, 0, stream>>>(rowp, eid, dst, lg, vbuf, x,
                                                     hup, disp, N);

    // 5) h_out = h + h_update @ Wo + bo   (WMMA, residual epilogue)
    gemm_wmma_f32<<<rowTiles, TB, 0, stream>>>(hup, WoT, bo, h, h_out, 0);
    // 6) t = SiLU(h_out @ Wg1 + bg1)      (WMMA, SiLU epilogue; t aliases q)
    gemm_wmma_f32<<<rowTiles, TB, 0, stream>>>(h_out, Wg1T, bg1, nullptr, t, 1);
    // 7) gate + coordinate update
    gate_coords_kernel<<<(N + TB - 1) / TB, TB, 0, stream>>>(t, wg2, bg2, x, disp, x_out, N);
}